// TensorCircuit_76776835383329
// MI455X (gfx1250) — compile-verified
//
#include <hip/hip_runtime.h>
#include <math.h>

#define QN 20
#define LAYERS 8
#define THREADS 512                       // 16 waves (wave32)
#define TILE_AMPS 4096u                   // 4096 complex64 per workgroup tile
#define NSTATES (1u << QN)
#define NBLOCKS (NSTATES / TILE_AMPS)     // 256 workgroups
#define APT (TILE_AMPS / THREADS)         // 8 amplitudes per thread

typedef __attribute__((ext_vector_type(2))) float v2f;
typedef __attribute__((ext_vector_type(8))) float v8f;

// Padded LDS indexing: +1 float2 per 16 amplitudes -> stride-16 column access
// (the pl=0 group) lands on distinct bank pairs instead of one.
__device__ __forceinline__ unsigned lidx(unsigned t) { return t + (t >> 4); }
#define TILE_LDS (TILE_AMPS + TILE_AMPS / 16)        // 4352 float2 = 34 KB

// ---------------------------------------------------------------------------
// CDNA5 async global->LDS gather (ASYNCcnt), saddr form: mem = SGPR64 + VGPR32.
// VDST VGPR holds the LDS byte address (low 32 bits of a generic LDS pointer).
// ---------------------------------------------------------------------------
__device__ __forceinline__ void async_load_b64(unsigned lds_addr, unsigned goff,
                                               unsigned long long gbase) {
  asm volatile("global_load_async_to_lds_b64 %0, %1, %2"
               :: "v"(lds_addr), "v"(goff), "s"(gbase) : "memory");
}
__device__ __forceinline__ void wait_async0() {
  asm volatile("s_wait_asynccnt 0" ::: "memory");
}

// ---------------------------------------------------------------------------
// Build the fused 4-qubit rotation U = R(q(pg+3)) (x) ... (x) R(q(pg)) as a
// dense 16x16 (row-major [m][n]); global bit b <-> qubit 19-b; group-value
// bit j <-> global bit pg+j.  R = [[c,-s],[s,c]], c/s of theta/2.
// ---------------------------------------------------------------------------
__device__ __forceinline__ void build_U(float* U, const float* theta,
                                        int layer, int pg) {
  for (unsigned e = threadIdx.x; e < 256u; e += blockDim.x) {
    unsigned m = e >> 4, n = e & 15u;
    float prod = 1.0f;
#pragma unroll
    for (int j = 0; j < 4; ++j) {
      int q = 19 - (pg + j);
      float th = 0.5f * theta[layer * QN + q];
      float cj = __cosf(th), sj = __sinf(th);
      unsigned mb = (m >> j) & 1u, nb = (n >> j) & 1u;
      prod *= (mb == nb) ? cj : ((mb == 0u) ? -sj : sj);
    }
    U[e] = prod;
  }
}

// ---------------------------------------------------------------------------
// Apply U (16x16) to the LDS tile on the nibble at local bit offset pl:
// tile index t = (hi << (pl+4)) | (g << pl) | lo, column c = (hi << pl)|lo.
// 256 columns -> 16 column-tiles; wave w owns columns [16w,16w+16) so passes
// are race-free.  D = sum_{kc=0..3} A_kc (16x4) * B_kc (4x16) via
// V_WMMA_F32_16X16X4_F32, once for re and once for im (U is real).
// ---------------------------------------------------------------------------
__device__ __forceinline__ void group_matmul(float2* tile, const float* U, int pl) {
  const unsigned tid   = threadIdx.x;
  const unsigned lane  = tid & 31u;
  const unsigned wave  = tid >> 5;
  const unsigned n     = lane & 15u;
  const unsigned khalf = (lane >> 4) << 1;            // 0 | 2 (ISA A/B half-lane split)
  const unsigned c     = wave * 16u + n;              // this lane's column
  const unsigned lo    = c & ((1u << pl) - 1u);
  const unsigned hi    = c >> pl;
  const unsigned base  = (hi << (pl + 4)) | lo;

  v2f a[4], bre[4], bim[4];
#pragma unroll
  for (int kc = 0; kc < 4; ++kc) {
    unsigned k0 = 4u * (unsigned)kc + khalf;
    a[kc].x = U[n * 16u + k0];                        // A: M = lane&15, K = k0
    a[kc].y = U[n * 16u + k0 + 1u];
    float2 x0 = tile[lidx(base | (k0 << pl))];        // B: rows k0,k0+1 of column c
    float2 x1 = tile[lidx(base | ((k0 + 1u) << pl))];
    bre[kc].x = x0.x; bre[kc].y = x1.x;
    bim[kc].x = x0.y; bim[kc].y = x1.y;
  }

  v8f dre = {}, dimg = {};
#pragma unroll
  for (int kc = 0; kc < 4; ++kc) {
    dre  = __builtin_amdgcn_wmma_f32_16x16x4_f32(false, a[kc], false, bre[kc],
                                                 (short)0, dre,  false, false);
    dimg = __builtin_amdgcn_wmma_f32_16x16x4_f32(false, a[kc], false, bim[kc],
                                                 (short)0, dimg, false, false);
  }

  const unsigned mbase = (lane >> 4) << 3;            // D: M = r + 8*(lane>=16)
#pragma unroll
  for (int r = 0; r < 8; ++r) {
    unsigned m = mbase + (unsigned)r;
    float2 o; o.x = dre[r]; o.y = dimg[r];
    tile[lidx(base | (m << pl))] = o;
  }
  __syncthreads();
}

// ---------------------------------------------------------------------------
__global__ void init_state_kernel(float2* __restrict__ psi) {
  unsigned i = blockIdx.x * blockDim.x + threadIdx.x;
  float2 v; v.x = (i == 0u) ? 1.0f : 0.0f; v.y = 0.0f;
  psi[i] = v;
}

// ---------------------------------------------------------------------------
// High kernel: qubit groups on global bits [12,16) and [16,20).
// Tile owns bits {19..12} U {3..0}: t = (h8<<4)|l4, i = (h8<<12)|(bb<<4)|l4.
// Global bit 12+j -> local bit 4+j, 16+j -> 8+j.
// ---------------------------------------------------------------------------
__global__ void __launch_bounds__(THREADS)
layer_high_kernel(float2* __restrict__ psi, const float* __restrict__ theta, int layer) {
  extern __shared__ float2 tile[];
  __shared__ float U[2][256];
  const unsigned tid = threadIdx.x;
  const unsigned bb  = blockIdx.x;                    // global bits 11..4
  const unsigned long long gbase = (unsigned long long)(uintptr_t)psi;

#pragma unroll
  for (int k = 0; k < (int)APT; ++k) {
    unsigned t  = tid + (unsigned)k * THREADS;
    unsigned i  = ((t >> 4) << 12) | (bb << 4) | (t & 15u);
    async_load_b64((unsigned)(uintptr_t)&tile[lidx(t)],
                   i * (unsigned)sizeof(float2), gbase);
  }
  build_U(U[0], theta, layer, 12);
  build_U(U[1], theta, layer, 16);
  wait_async0();
  __syncthreads();

  group_matmul(tile, U[0], 4);
  group_matmul(tile, U[1], 8);

#pragma unroll
  for (int k = 0; k < (int)APT; ++k) {
    unsigned t = tid + (unsigned)k * THREADS;
    unsigned i = ((t >> 4) << 12) | (bb << 4) | (t & 15u);
    psi[i] = tile[lidx(t)];
  }
}

// ---------------------------------------------------------------------------
// Low kernel: qubit groups on global bits [0,4),[4,8),[8,12) on a contiguous
// 4096-amp tile, then the fused diagonal:  CZ sign = (-1)^popc(i & i>>1),
// Rzz angle = S - 2*sum_{gray-bit p set} h[18-p], h_q = -phi_q/2,
// gray = i ^ (i>>1).  Final layer: diagonal is measurement-invariant, so we
// skip it and emit float probabilities to d_out.
// ---------------------------------------------------------------------------
__global__ void __launch_bounds__(THREADS)
layer_low_kernel(float2* __restrict__ psi, const float* __restrict__ theta,
                 const float* __restrict__ phi, int layer, float* __restrict__ probs) {
  extern __shared__ float2 tile[];
  __shared__ float U[3][256];
  __shared__ float hph[QN - 1];
  const unsigned tid = threadIdx.x;
  const unsigned blk = blockIdx.x;                    // global bits 19..12
  const unsigned long long gbase =
      (unsigned long long)(uintptr_t)psi +
      (unsigned long long)blk * TILE_AMPS * sizeof(float2);

#pragma unroll
  for (int k = 0; k < (int)APT; ++k) {
    unsigned t = tid + (unsigned)k * THREADS;
    async_load_b64((unsigned)(uintptr_t)&tile[lidx(t)],
                   t * (unsigned)sizeof(float2), gbase);
  }
  build_U(U[0], theta, layer, 0);
  build_U(U[1], theta, layer, 4);
  build_U(U[2], theta, layer, 8);
  if (probs == nullptr && tid < QN - 1)
    hph[tid] = -0.5f * phi[layer * (QN - 1) + tid];
  wait_async0();
  __syncthreads();

  group_matmul(tile, U[0], 0);
  group_matmul(tile, U[1], 4);
  group_matmul(tile, U[2], 8);

  if (probs != nullptr) {
#pragma unroll
    for (int k = 0; k < (int)APT; ++k) {
      unsigned t = tid + (unsigned)k * THREADS;
      float2 a = tile[lidx(t)];
      probs[(blk << 12) | t] = a.x * a.x + a.y * a.y;
    }
  } else {
    float S = 0.0f;
    for (int q = 0; q < QN - 1; ++q) S += hph[q];
#pragma unroll
    for (int k = 0; k < (int)APT; ++k) {
      unsigned t    = tid + (unsigned)k * THREADS;
      unsigned i    = (blk << 12) | t;
      unsigned gray = i ^ (i >> 1);
      float acc = 0.0f;
      for (int p = 0; p < QN - 1; ++p)
        if ((gray >> p) & 1u) acc += hph[(QN - 2) - p];
      float A   = S - 2.0f * acc;
      float sgn = (__popc(i & (i >> 1)) & 1) ? -1.0f : 1.0f;
      float sa, ca;
      __sincosf(A, &sa, &ca);
      ca *= sgn; sa *= sgn;
      float2 a = tile[lidx(t)];
      float2 r;
      r.x = a.x * ca - a.y * sa;
      r.y = a.x * sa + a.y * ca;
      psi[i] = r;
    }
  }
}

// ---------------------------------------------------------------------------
// 17 passes over an 8 MB state that never leaves the 192 MB L2; 20 Ry gates
// per layer applied as five 16x16 WMMA matmul passes; diagonals fused.
// ---------------------------------------------------------------------------
extern "C" void kernel_launch(void* const* d_in, const int* in_sizes, int n_in,
                              void* d_out, int out_size, void* d_ws, size_t ws_size,
                              hipStream_t stream) {
  (void)in_sizes; (void)n_in; (void)out_size; (void)ws_size;
  const float* theta = (const float*)d_in[0];   // [LAYERS][QN]
  const float* phi   = (const float*)d_in[1];   // [LAYERS][QN-1]
  float2* psi = (float2*)d_ws;                  // 2^20 complex64 = 8 MB
  float*  out = (float*)d_out;                  // 2^20 probabilities

  init_state_kernel<<<NSTATES / 256, 256, 0, stream>>>(psi);

  const size_t lds_bytes = (size_t)TILE_LDS * sizeof(float2);  // ~34 KB
  for (int l = 0; l < LAYERS; ++l) {
    layer_high_kernel<<<NBLOCKS, THREADS, lds_bytes, stream>>>(psi, theta, l);
    layer_low_kernel<<<NBLOCKS, THREADS, lds_bytes, stream>>>(
        psi, theta, phi, l, (l == LAYERS - 1) ? out : nullptr);
  }
}